// SpatialKDEConv_23519240913319
// MI455X (gfx1250) — compile-verified
//
#include <hip/hip_runtime.h>
#include <cstdint>

// ---------------------------------------------------------------------------
// SpatialKDEConv for MI455X (gfx1250).
// B=64, G=64, A=8400 (80x80 + 40x40 + 20x20). Sparse-scatter formulation:
// the "depthwise conv" input has <=10 nonzeros per (b,g) row, so conv ==
// scatter of 10 x 49 weighted taps. Memory-bound: ~310MB traffic -> ~13us
// at 23.3 TB/s. CDNA5 paths used:
//   * V_WMMA_F32_16X16X4_F32 : rank-1 outer product inv_bw[16] x logK[16]
//     (the per-GT Gaussian kernel exponents), then exp() in VALU.
//   * global_load_async_to_lds_b128 + s_wait_asynccnt : stage each 33.6KB
//     align_metric row into LDS for the iterative top-10 search.
//   * ds_add_f32 (atomicAdd on LDS) for the tap scatter.
// ---------------------------------------------------------------------------

#define NB 64
#define NG 64
#define NPAIR (NB * NG)
#define NA 8400
#define L0N 6400          // 80x80
#define L1N 1600          // 40x40
#define KK 49             // 7x7
#define MAXK 10
#define MINK 4
#define THR 3.0f

typedef __attribute__((ext_vector_type(2))) float v2f;
typedef __attribute__((ext_vector_type(8))) float v8f;

__device__ __forceinline__ void wait_async_zero() {
#if __has_builtin(__builtin_amdgcn_s_wait_asynccnt)
  __builtin_amdgcn_s_wait_asynccnt(0);
#else
  asm volatile("s_wait_asynccnt 0" ::: "memory");
#endif
}

// ---------------------------------------------------------------------------
// Stage K: per-pair 7x7 kernels via WMMA outer product.
// kern(p, t) = exp( (1/bw_p) * logKbase_t ). One wave handles 16 pairs (M)
// and 16 taps (N) per WMMA; K-dim = 1 (only K=0 lanes populated).
// Grid: 4096 pairs / 16 per wave = 256 waves = 64 blocks x 128 threads.
// ---------------------------------------------------------------------------
__global__ __launch_bounds__(128) void kde_kernels_wmma(
    const float* __restrict__ gt_boxes,          // (NPAIR, 4) xyxy
    const unsigned char* __restrict__ mask_gt,   // (NPAIR,)
    float* __restrict__ kerns) {                 // (NPAIR, 49)
  int tid = blockIdx.x * blockDim.x + threadIdx.x;
  int lane = tid & 31;
  int wave = tid >> 5;
  int pairBase = wave * 16;
  int laneN = lane & 15;
  bool lowHalf = lane < 16;

  // A operand: A[m][k=0] = inv_bw for pair (pairBase+m); all other K = 0.
  int p = pairBase + laneN;                      // grid exact: p < NPAIR
  float x0 = gt_boxes[p * 4 + 0];
  float y0 = gt_boxes[p * 4 + 1];
  float x1 = gt_boxes[p * 4 + 2];
  float y1 = gt_boxes[p * 4 + 3];
  float bw = 0.01f * sqrtf((x1 - x0) * (y1 - y0) + 1e-9f);
  bw = (mask_gt[p] != 0) ? bw : 1.0f;            // invalid rows -> safe
  float inv = 1.0f / bw;
  v2f a;
  a.x = lowHalf ? inv : 0.0f;                    // lanes 16..31 carry K=2 -> 0
  a.y = 0.0f;                                    // K=1 / K=3 -> 0

  // log of the normalized base Gaussian: log k(dy,dx) = -(dx^2+dy^2)/2 - log(sum)
  float S = 0.0f;
#pragma unroll
  for (int i = -3; i <= 3; ++i) S += expf(-0.5f * (float)(i * i));
  float logKsum = 2.0f * logf(S);                // separable: sum = S^2

#pragma unroll
  for (int g = 0; g < 4; ++g) {                  // 49 taps in 4 groups of 16
    int tap = g * 16 + laneN;
    int dy = tap / 7 - 3;
    int dx = tap % 7 - 3;
    float lk = (tap < KK) ? (-0.5f * (float)(dx * dx + dy * dy) - logKsum) : 0.0f;
    v2f b;
    b.x = lowHalf ? lk : 0.0f;                   // only K=0 row of B
    b.y = 0.0f;
    v8f c = {0.f, 0.f, 0.f, 0.f, 0.f, 0.f, 0.f, 0.f};
    // D[m][n] = inv_bw[m] * logK[n]   (emits v_wmma_f32_16x16x4_f32)
    v8f d = __builtin_amdgcn_wmma_f32_16x16x4_f32(
        /*neg_a=*/false, a, /*neg_b=*/false, b,
        /*c_mod=*/(short)0, c, /*reuse_a=*/false, /*reuse_b=*/false);
#pragma unroll
    for (int i = 0; i < 8; ++i) {
      int m = lowHalf ? i : (8 + i);             // C/D VGPR layout
      int t = g * 16 + laneN;
      if (t < KK) kerns[(pairBase + m) * KK + t] = expf(d[i]);
    }
  }
}

// ---------------------------------------------------------------------------
// Stage T: top-10 + dynamic-k per pair. One 256-thread block per pair.
// Row (8400 f32 = 33.6KB, 2100 x b128) staged into LDS with async loads.
// ---------------------------------------------------------------------------
__global__ __launch_bounds__(256) void kde_topk(
    const float* __restrict__ am,                // (NPAIR, NA)
    const unsigned char* __restrict__ mask_gt,   // (NPAIR,)
    float* __restrict__ sel_val,                 // (NPAIR, MAXK)
    int* __restrict__ sel_idx) {                 // (NPAIR, MAXK)
  __shared__ __align__(16) float sdata[NA];
  __shared__ float smax[256];
  __shared__ int sidx[256];
  __shared__ float topv[MAXK];
  __shared__ int topi[MAXK];
  __shared__ int sk;

  int pair = blockIdx.x;
  int tid = threadIdx.x;

  // --- async stage row -> LDS (ASYNCcnt path) ---
  uint32_t ldsBase = (uint32_t)(uintptr_t)&sdata[0];    // low 32 bits = LDS byte addr
  uint64_t base = (uint64_t)(uintptr_t)am;
  uint32_t rowOff = (uint32_t)pair * (uint32_t)(NA * 4);
  for (int i = tid; i < NA / 4; i += 256) {             // 2100 b128 transfers
    uint32_t l = ldsBase + (uint32_t)i * 16u;
    uint32_t g = rowOff + (uint32_t)i * 16u;
    asm volatile("global_load_async_to_lds_b128 %0, %1, %2"
                 :: "v"(l), "v"(g), "s"(base)
                 : "memory");
  }
  wait_async_zero();
  __syncthreads();

  // --- 10 iterative arg-max passes (ties -> lower index, matching lax.top_k) ---
  for (int pass = 0; pass < MAXK; ++pass) {
    float bv = -1.0f;                                    // metrics in [0,1)
    int bi = 0x7fffffff;
    for (int a2 = tid; a2 < NA; a2 += 256) {
      float v = sdata[a2];
      if (v > bv || (v == bv && a2 < bi)) { bv = v; bi = a2; }
    }
    smax[tid] = bv;
    sidx[tid] = bi;
    __syncthreads();
    for (int s = 128; s > 0; s >>= 1) {
      if (tid < s) {
        float ov = smax[tid + s];
        int oi = sidx[tid + s];
        if (ov > smax[tid] || (ov == smax[tid] && oi < sidx[tid])) {
          smax[tid] = ov;
          sidx[tid] = oi;
        }
      }
      __syncthreads();
    }
    if (tid == 0) {
      topv[pass] = smax[0];
      topi[pass] = sidx[0];
      sdata[sidx[0]] = -1.0f;                            // knock out winner
    }
    __syncthreads();
  }

  // --- dynamic k: first index where cumsum > 3.0, clipped to [4,10]; gate mask_gt ---
  if (tid == 0) {
    float cs = 0.0f;
    int k = MAXK;
    for (int j = 0; j < MAXK; ++j) {
      cs += topv[j];
      if (cs > THR) { k = j; break; }
    }
    if (k < MINK) k = MINK;
    if (k > MAXK) k = MAXK;
    sk = (mask_gt[pair] != 0) ? k : 0;
  }
  __syncthreads();
  if (tid < MAXK) {
    bool on = tid < sk;
    sel_val[pair * MAXK + tid] = on ? topv[tid] : 0.0f;
    sel_idx[pair * MAXK + tid] = on ? topi[tid] : 0;
  }
}

// ---------------------------------------------------------------------------
// Stage S: sparse scatter-conv + mask_in_gts + row-max normalize.
// One 256-thread block per pair; output row built in LDS.
// ---------------------------------------------------------------------------
__global__ __launch_bounds__(256) void kde_scatter(
    const float* __restrict__ sel_val,
    const int* __restrict__ sel_idx,
    const float* __restrict__ kerns,
    const unsigned char* __restrict__ mig,       // (NPAIR, NA) mask_in_gts
    float* __restrict__ out) {                   // (NPAIR, NA)
  __shared__ float srow[NA];
  __shared__ float kk_[KK];
  __shared__ float sv[MAXK];
  __shared__ int si[MAXK];
  __shared__ float smax[256];

  int pair = blockIdx.x;
  int tid = threadIdx.x;

  for (int a = tid; a < NA; a += 256) srow[a] = 0.0f;
  if (tid < KK) kk_[tid] = kerns[pair * KK + tid];
  if (tid < MAXK) {
    sv[tid] = sel_val[pair * MAXK + tid];
    si[tid] = sel_idx[pair * MAXK + tid];
  }
  __syncthreads();

  int dy = tid / 7 - 3;                          // valid for tid < 49
  int dx = tid % 7 - 3;
  for (int j = 0; j < MAXK; ++j) {               // uniform loop over selections
    float val = sv[j];
    if (val != 0.0f && tid < KK) {
      int idx = si[j];
      int w, off;
      if (idx < L0N)            { w = 80; off = 0; }
      else if (idx < L0N + L1N) { w = 40; off = L0N; }
      else                      { w = 20; off = L0N + L1N; }
      int loc = idx - off;
      int y = loc / w, x = loc % w;
      int yy = y + dy, xx = x + dx;
      if (yy >= 0 && yy < w && xx >= 0 && xx < w)          // zero-pad border
        atomicAdd(&srow[off + yy * w + xx], val * kk_[tid]); // ds_add_f32
    }
  }
  __syncthreads();

  // mask_in_gts, then row max (invalid/empty rows stay all-zero -> max 0)
  float lm = 0.0f;
  size_t rowBase = (size_t)pair * (size_t)NA;
  for (int a = tid; a < NA; a += 256) {
    float m = srow[a] * ((mig[rowBase + a] != 0) ? 1.0f : 0.0f);
    srow[a] = m;
    lm = fmaxf(lm, m);
  }
  smax[tid] = lm;
  __syncthreads();
  for (int s = 128; s > 0; s >>= 1) {
    if (tid < s) smax[tid] = fmaxf(smax[tid], smax[tid + s]);
    __syncthreads();
  }
  float rm = smax[0];
  float scale = (rm > 0.0f) ? (1.0f / (rm + 1e-9f)) : 1.0f;
  for (int a = tid; a < NA; a += 256) out[rowBase + a] = srow[a] * scale;
}

// ---------------------------------------------------------------------------
extern "C" void kernel_launch(void* const* d_in, const int* in_sizes, int n_in,
                              void* d_out, int out_size, void* d_ws, size_t ws_size,
                              hipStream_t stream) {
  const float* am = (const float*)d_in[0];              // align_metric (B,G,A) f32
  const float* gt = (const float*)d_in[1];              // gt_boxes (B,G,4) f32
  const unsigned char* mg = (const unsigned char*)d_in[2];   // mask_gt (B,G,1) bool
  const unsigned char* mig = (const unsigned char*)d_in[3];  // mask_in_gts (B,G,A) bool
  float* out = (float*)d_out;

  char* ws = (char*)d_ws;                               // ~1.13 MB, rewritten each call
  float* kerns = (float*)ws;                                            // NPAIR*49 f32
  float* sel_val = (float*)(ws + (size_t)NPAIR * KK * sizeof(float));   // NPAIR*10 f32
  int* sel_idx = (int*)(ws + (size_t)NPAIR * KK * sizeof(float)
                           + (size_t)NPAIR * MAXK * sizeof(float));     // NPAIR*10 i32

  // 4096 pairs / 16 per wave = 256 waves; 4 waves per 128-thread block.
  kde_kernels_wmma<<<NPAIR / 16 / 4, 128, 0, stream>>>(gt, mg, kerns);
  kde_topk<<<NPAIR, 256, 0, stream>>>(am, mg, sel_val, sel_idx);
  kde_scatter<<<NPAIR, 256, 0, stream>>>(sel_val, sel_idx, kerns, mig, out);
}